// GCN_4930622456147
// MI455X (gfx1250) — compile-verified
//
#include <hip/hip_runtime.h>

typedef float v2f __attribute__((ext_vector_type(2)));
typedef float v8f __attribute__((ext_vector_type(8)));

static constexpr int F_IN  = 128;
static constexpr int F_HID = 16;
static constexpr int F_OUT = 8;

// ---------------------------------------------------------------------------
// Degree / GCN normalization
// ---------------------------------------------------------------------------
__global__ void gcn_deg_init(float* __restrict__ deg, int n) {
  int i = blockIdx.x * blockDim.x + threadIdx.x;
  if (i < n) deg[i] = 1.0f;                    // self-loop contributes 1
}

__global__ void gcn_deg_accum(const int* __restrict__ dst, float* __restrict__ deg, int e) {
  int i = blockIdx.x * blockDim.x + threadIdx.x;
  if (i < e) atomicAdd(&deg[dst[i]], 1.0f);    // L2-resident f32 atomic
}

__global__ void gcn_dinv(const float* __restrict__ deg, float* __restrict__ dinv, int n) {
  int i = blockIdx.x * blockDim.x + threadIdx.x;
  if (i < n) dinv[i] = rsqrtf(deg[i]);         // deg >= 1 always (self-loops)
}

// ---------------------------------------------------------------------------
// Layer-1 transform: h1[N,16] = x[N,128] @ W1[128,16]  via v_wmma_f32_16x16x4_f32
// One wave computes one 16-row tile; K=128 -> 32 chained WMMAs.
// A layout (f32 16x4): lane l: m=l&15, k = 2*(l>>4)+{0,1} in {a.x,a.y}
// B layout (f32 4x16): lane l: n=l&15, same k split
// C/D layout: VGPR r: M = r + 8*(l>>4), N = l&15
// ---------------------------------------------------------------------------
__global__ __launch_bounds__(256) void gcn_gemm1_wmma(
    const float* __restrict__ x, const float* __restrict__ W1,
    float* __restrict__ h1, int n) {
  const int lane = threadIdx.x & 31;
  const int wave = (int)((blockIdx.x * blockDim.x + threadIdx.x) >> 5);
  const int ntiles = n >> 4;
  if (wave >= ntiles) return;                  // wave-uniform: EXEC all-1 at WMMA
  const int row0 = wave << 4;
  const int half = lane >> 4;                  // 0 | 1
  const int m    = lane & 15;                  // row (A) / col (B,D)
  const int kh   = half << 1;                  // 0 | 2

  const float* arow = x  + (size_t)(row0 + m) * F_IN + kh;
  const float* bcol = W1 + (size_t)kh * F_HID + m;

  v8f c = {0.f, 0.f, 0.f, 0.f, 0.f, 0.f, 0.f, 0.f};
#pragma unroll 4
  for (int k = 0; k < F_IN; k += 4) {
    v2f a, b;
    a.x = arow[k];
    a.y = arow[k + 1];
    b.x = bcol[(size_t)k * F_HID];
    b.y = bcol[(size_t)(k + 1) * F_HID];
    c = __builtin_amdgcn_wmma_f32_16x16x4_f32(false, a, false, b,
                                              (short)0, c, false, false);
  }
  float* orow = h1 + (size_t)(row0 + (half << 3)) * F_HID + m;
#pragma unroll
  for (int r = 0; r < 8; ++r) orow[(size_t)r * F_HID] = c[r];
}

// ---------------------------------------------------------------------------
// Layer-2 transform: h2[N,8] = relu(a1[N,16]) @ W2[16,8], W2 zero-padded to 16
// cols so a single 16x16 WMMA tile covers it. ReLU fused into the A load.
// ---------------------------------------------------------------------------
__global__ __launch_bounds__(256) void gcn_gemm2_wmma(
    const float* __restrict__ a1, const float* __restrict__ W2,
    float* __restrict__ h2, int n) {
  const int lane = threadIdx.x & 31;
  const int wave = (int)((blockIdx.x * blockDim.x + threadIdx.x) >> 5);
  const int ntiles = n >> 4;
  if (wave >= ntiles) return;
  const int row0 = wave << 4;
  const int half = lane >> 4;
  const int m    = lane & 15;
  const int kh   = half << 1;

  const float* arow = a1 + (size_t)(row0 + m) * F_HID + kh;

  v8f c = {0.f, 0.f, 0.f, 0.f, 0.f, 0.f, 0.f, 0.f};
#pragma unroll
  for (int k = 0; k < F_HID; k += 4) {
    v2f a, b;
    a.x = fmaxf(arow[k],     0.0f);            // fused ReLU
    a.y = fmaxf(arow[k + 1], 0.0f);
    float b0 = 0.0f, b1 = 0.0f;
    if (m < F_OUT) {                           // zero-pad cols 8..15 of W2
      b0 = W2[(size_t)(k + kh)     * F_OUT + m];
      b1 = W2[(size_t)(k + kh + 1) * F_OUT + m];
    }
    b.x = b0; b.y = b1;
    c = __builtin_amdgcn_wmma_f32_16x16x4_f32(false, a, false, b,
                                              (short)0, c, false, false);
  }
  if (m < F_OUT) {
    float* orow = h2 + (size_t)(row0 + (half << 3)) * F_OUT + m;
#pragma unroll
    for (int r = 0; r < 8; ++r) orow[(size_t)r * F_OUT] = c[r];
  }
}

// ---------------------------------------------------------------------------
// Aggregation: out[i,f] = bias[f] + dinv[i]^2 * h[i,f]   (self-loop + init)
//              out[d,f] += dinv[s]*dinv[d] * h[s,f]      (edge scatter-add)
// Lane-per-feature: F contiguous lanes share one edge -> coalesced 64B/32B
// gathers and atomics; h/out arrays (6.4MB) are L2-resident on MI455X.
// ---------------------------------------------------------------------------
template <int F>
__global__ void gcn_agg_init(const float* __restrict__ h,
                             const float* __restrict__ dinv,
                             const float* __restrict__ bias,
                             float* __restrict__ out, int n) {
  int t = blockIdx.x * blockDim.x + threadIdx.x;
  if (t >= n * F) return;
  int i = t / F;
  int j = t - i * F;
  float di = dinv[i];
  out[t] = bias[j] + di * di * h[t];
}

template <int F>
__global__ void gcn_agg_edges(const int* __restrict__ src,
                              const int* __restrict__ dst,
                              const float* __restrict__ h,
                              const float* __restrict__ dinv,
                              float* __restrict__ out, int e) {
  int t = blockIdx.x * blockDim.x + threadIdx.x;
  if (t >= e * F) return;                      // max 51.2M, fits int32
  int ei = t / F;
  int j  = t - ei * F;
  int s = src[ei];
  int d = dst[ei];
  float norm = dinv[s] * dinv[d];
  atomicAdd(&out[(size_t)d * F + j], norm * h[(size_t)s * F + j]);
}

// ---------------------------------------------------------------------------
// log_softmax over 8 logits per node, in place on d_out.
// ---------------------------------------------------------------------------
__global__ void gcn_logsoftmax8(float* __restrict__ out, int n) {
  int i = blockIdx.x * blockDim.x + threadIdx.x;
  if (i >= n) return;
  float* p = out + (size_t)i * 8;
  float4 lo = *reinterpret_cast<const float4*>(p);
  float4 hi = *reinterpret_cast<const float4*>(p + 4);
  float v[8] = {lo.x, lo.y, lo.z, lo.w, hi.x, hi.y, hi.z, hi.w};
  float m = v[0];
#pragma unroll
  for (int j = 1; j < 8; ++j) m = fmaxf(m, v[j]);
  float s = 0.0f;
#pragma unroll
  for (int j = 0; j < 8; ++j) s += expf(v[j] - m);
  float l = m + logf(s);
  float4 olo = {v[0] - l, v[1] - l, v[2] - l, v[3] - l};
  float4 ohi = {v[4] - l, v[5] - l, v[6] - l, v[7] - l};
  *reinterpret_cast<float4*>(p)     = olo;
  *reinterpret_cast<float4*>(p + 4) = ohi;
}

// ---------------------------------------------------------------------------
// Launch
// ---------------------------------------------------------------------------
extern "C" void kernel_launch(void* const* d_in, const int* in_sizes, int n_in,
                              void* d_out, int out_size, void* d_ws, size_t ws_size,
                              hipStream_t stream) {
  const float* x   = (const float*)d_in[0];
  const int*   ei  = (const int*)  d_in[1];   // [2, E] row-major, int32
  const float* W1  = (const float*)d_in[2];
  const float* b1  = (const float*)d_in[3];
  const float* W2  = (const float*)d_in[4];
  const float* b2  = (const float*)d_in[5];
  float*       out = (float*)d_out;

  const int n = in_sizes[0] / F_IN;           // 100000
  const int e = in_sizes[1] / 2;              // 3200000
  const int* src = ei;
  const int* dst = ei + e;

  // Workspace layout (floats): deg | dinv | h1[N,16] | a1[N,16] | h2[N,8]
  float* ws   = (float*)d_ws;
  float* deg  = ws;                     ws += n;
  float* dinv = ws;                     ws += n;
  float* h1   = ws;                     ws += (size_t)n * F_HID;
  float* a1   = ws;                     ws += (size_t)n * F_HID;
  float* h2   = ws;                     // n * F_OUT

  const int B = 256;
  auto cdiv = [](long long a, long long b) { return (int)((a + b - 1) / b); };

  // norm coefficients
  gcn_deg_init <<<cdiv(n, B), B, 0, stream>>>(deg, n);
  gcn_deg_accum<<<cdiv(e, B), B, 0, stream>>>(dst, deg, e);
  gcn_dinv     <<<cdiv(n, B), B, 0, stream>>>(deg, dinv, n);

  // layer 1: transform (WMMA) -> aggregate
  const int waves = (n + 15) / 16;            // one wave per 16-row tile
  gcn_gemm1_wmma<<<cdiv((long long)waves * 32, B), B, 0, stream>>>(x, W1, h1, n);
  gcn_agg_init<F_HID> <<<cdiv((long long)n * F_HID, B), B, 0, stream>>>(h1, dinv, b1, a1, n);
  gcn_agg_edges<F_HID><<<cdiv((long long)e * F_HID, B), B, 0, stream>>>(src, dst, h1, dinv, a1, e);

  // layer 2: ReLU+transform (WMMA) -> aggregate into d_out
  gcn_gemm2_wmma<<<cdiv((long long)waves * 32, B), B, 0, stream>>>(a1, W2, h2, n);
  gcn_agg_init<F_OUT> <<<cdiv((long long)n * F_OUT, B), B, 0, stream>>>(h2, dinv, b2, out, n);
  gcn_agg_edges<F_OUT><<<cdiv((long long)e * F_OUT, B), B, 0, stream>>>(src, dst, h2, dinv, out, e);

  // epilogue
  gcn_logsoftmax8<<<cdiv(n, B), B, 0, stream>>>(out, n);
}